// VectorQuantizer_46196668236383
// MI455X (gfx1250) — compile-verified
//
#include <hip/hip_runtime.h>

// ---------------------------------------------------------------------------
// VQ-VAE vector quantizer for MI455X (gfx1250, wave32).
//   X:      [16, 64, 64, 64] f32 (NCHW)  -> z_e rows n = b*4096 + h*64 + w
//   weight: [1024, 64] f32 codebook
// Outputs (concatenated in d_out):
//   z_q:     [16, 64, 64, 64] f32  (4,194,304 floats)
//   one_hot: [65536, 1024] f32     (67,108,864 floats)
// GEMM via V_WMMA_F32_16X16X4_F32 (exact f32 matrix path).
// ---------------------------------------------------------------------------

typedef float v2f __attribute__((ext_vector_type(2)));
typedef float v8f __attribute__((ext_vector_type(8)));

#define CH          64
#define HW          4096          // 64*64
#define NROWS       65536         // 16*4096
#define KCODES      1024
#define DDIM        64
#define TILE_K      16            // codes per LDS tile
#define ROWS_PER_WAVE   16
#define WAVES_PER_BLOCK 8
#define ROWS_PER_BLOCK  128
#define LDS_PITCH   66            // padded row pitch (floats) to spread banks

// --- kernel 1: zero the one-hot region (256 MB) ----------------------------
__global__ void vq_zero_onehot(float4* __restrict__ p, long n4) {
    long i = (long)blockIdx.x * blockDim.x + threadIdx.x;
    long stride = (long)gridDim.x * blockDim.x;
    float4 z = make_float4(0.f, 0.f, 0.f, 0.f);
    for (; i < n4; i += stride) p[i] = z;
}

// --- kernel 2: per-code squared norms --------------------------------------
__global__ void vq_code_norms(const float* __restrict__ w,
                              float* __restrict__ norms) {
    int k = blockIdx.x * blockDim.x + threadIdx.x;
    if (k >= KCODES) return;
    const float* r = w + (size_t)k * DDIM;
    float s = 0.f;
#pragma unroll
    for (int c = 0; c < DDIM; ++c) s = fmaf(r[c], r[c], s);
    norms[k] = s;
}

// --- kernel 3: distances via WMMA f32, argmin, scatter ----------------------
__global__ void __launch_bounds__(256)
vq_main(const float* __restrict__ X, const float* __restrict__ W,
        const float* __restrict__ norms, float* __restrict__ zq,
        float* __restrict__ onehot) {
    __shared__ float sW[TILE_K * LDS_PITCH];                 // 4224 B
    __shared__ int   sIdx[WAVES_PER_BLOCK * ROWS_PER_WAVE];  // 512 B

    const int tid  = threadIdx.x;
    const int wid  = tid >> 5;
    const int lane = tid & 31;
    const int l    = lane & 15;        // column / row-in-tile coordinate
    const int half = lane >> 4;        // 0: lanes 0-15, 1: lanes 16-31

    const int row_base = blockIdx.x * ROWS_PER_BLOCK + wid * ROWS_PER_WAVE;
    const int b_img = row_base >> 12;      // row / 4096
    const int p0    = row_base & 4095;     // pixel offset within image
    const float* xbase = X + (size_t)b_img * (CH * HW) + p0;

    // Preload A fragments: 16 k-steps of a 16x4 f32 tile.
    // ISA layout: VGPR0 = {K=0 | K=2}, VGPR1 = {K=1 | K=3} across half-waves.
    v2f af[16];
#pragma unroll
    for (int s = 0; s < 16; ++s) {
        int c0 = 4 * s + 2 * half;
        af[s].x = xbase[(size_t)c0 * HW + l];          // z_e[row_base+l, c0]
        af[s].y = xbase[(size_t)(c0 + 1) * HW + l];    // z_e[row_base+l, c0+1]
    }

    float best[8];
    int   bidx[8];
#pragma unroll
    for (int m = 0; m < 8; ++m) { best[m] = 3.0e38f; bidx[m] = 0x7fffffff; }

    for (int t = 0; t < KCODES / TILE_K; ++t) {
        const int code_base = t * TILE_K;
        __syncthreads();
        // Cooperatively stage 16 codes x 64 dims (4 KB) into padded LDS.
#pragma unroll
        for (int i = 0; i < 4; ++i) {
            int j = i * 256 + tid;
            int row = j >> 6, col = j & 63;
            sW[row * LDS_PITCH + col] = W[(size_t)code_base * DDIM + j];
        }
        if (t + 1 < KCODES / TILE_K)   // hint next tile -> global_prefetch_b8
            __builtin_prefetch(&W[(size_t)(code_base + TILE_K) * DDIM + tid * 4], 0, 1);
        __syncthreads();

        // dot(z_e tile, code tile): 16 WMMA f32 16x16x4 steps over D=64.
        v8f acc = {0.f, 0.f, 0.f, 0.f, 0.f, 0.f, 0.f, 0.f};
#pragma unroll
        for (int s = 0; s < 16; ++s) {
            int k0 = 4 * s + 2 * half;   // B mirrors A: VGPR0={K0|K2}, VGPR1={K1|K3}
            v2f b;
            b.x = sW[l * LDS_PITCH + k0];        // B[k, n=l] = weight[code l][k0+k]
            b.y = sW[l * LDS_PITCH + k0 + 1];
            acc = __builtin_amdgcn_wmma_f32_16x16x4_f32(
                /*neg_a=*/false, af[s], /*neg_b=*/false, b,
                /*c_mod=*/(short)0, acc, /*reuse_a=*/false, /*reuse_b=*/false);
        }

        // score = ||e||^2 - 2*dot  (||z||^2 constant per row, drops from argmin)
        const float nrm  = norms[code_base + l];
        const int   code = code_base + l;
#pragma unroll
        for (int m = 0; m < 8; ++m) {
            float sc = fmaf(-2.f, acc[m], nrm);
            if (sc < best[m]) { best[m] = sc; bidx[m] = code; }  // keeps lowest idx on tie
        }
    }

    // Cross-lane argmin within each 16-lane half (C/D layout: half0 -> M=0..7,
    // half1 -> M=8..15). Tie-break on smaller index to match jnp.argmin.
#pragma unroll
    for (int off = 1; off < 16; off <<= 1) {
#pragma unroll
        for (int m = 0; m < 8; ++m) {
            float s2 = __shfl_xor(best[m], off, 32);
            int   i2 = __shfl_xor(bidx[m], off, 32);
            if (s2 < best[m] || (s2 == best[m] && i2 < bidx[m])) {
                best[m] = s2; bidx[m] = i2;
            }
        }
    }
    if (l == 0) {
#pragma unroll
        for (int m = 0; m < 8; ++m)
            sIdx[wid * 16 + half * 8 + m] = bidx[m];
    }
    __syncthreads();

    const int idx_r = sIdx[wid * 16 + l];   // winning code for row row_base + l

    // one-hot: one 1.0f per row (region pre-zeroed by vq_zero_onehot)
    if (half == 0)
        onehot[(size_t)(row_base + l) * KCODES + idx_r] = 1.0f;

    // z_q NCHW scatter: per channel, 16 consecutive pixels -> 64B segments.
    float* zb = zq + (size_t)b_img * (CH * HW) + p0;
#pragma unroll
    for (int c2 = 0; c2 < 32; ++c2) {
        int c = 2 * c2 + half;                       // half0: even ch, half1: odd ch
        zb[(size_t)c * HW + l] = W[(size_t)idx_r * DDIM + c];
    }
}

extern "C" void kernel_launch(void* const* d_in, const int* in_sizes, int n_in,
                              void* d_out, int out_size, void* d_ws, size_t ws_size,
                              hipStream_t stream) {
    (void)in_sizes; (void)n_in; (void)out_size; (void)ws_size;
    const float* X = (const float*)d_in[0];        // [16,64,64,64]
    const float* W = (const float*)d_in[1];        // [1024,64]
    float* zq     = (float*)d_out;                                   // 4,194,304 f32
    float* onehot = (float*)d_out + (size_t)NROWS * DDIM;            // 67,108,864 f32
    float* norms  = (float*)d_ws;                                    // 1024 f32

    // 1) zero one-hot region (16 MB offset keeps float4 alignment)
    long n4 = ((long)NROWS * KCODES) / 4;          // 16,777,216 float4
    vq_zero_onehot<<<2048, 256, 0, stream>>>((float4*)onehot, n4);

    // 2) code norms
    vq_code_norms<<<KCODES / 256, 256, 0, stream>>>(W, norms);

    // 3) WMMA distance + argmin + scatter
    vq_main<<<NROWS / ROWS_PER_BLOCK, 256, 0, stream>>>(X, W, norms, zq, onehot);
}